// VSSBlock_45243185496329
// MI455X (gfx1250) — compile-verified
//
#include <hip/hip_runtime.h>
#include <hip/hip_bf16.h>
#include <math.h>

// ---------------- problem constants ----------------
#define BB   4
#define DD   8
#define HH   32
#define WW   32
#define CC   192
#define DI   384          // d_inner
#define NS   16           // n_state
#define DR   12           // dt_rank
#define LL   (DD*HH*WW)   // 8192
#define BL   (BB*LL)      // 32768
#define NPAD 48           // padded x_proj rows (44 -> 48)
#define CH   128          // scan chunk length
#define NCH  (LL/CH)      // 64 chunks
#define EPSL 1e-5f

#define KC   192          // GEMM K-chunk staged in LDS
#define LP   194          // LDS pitch (floats): 194 % 64 == 2 -> conflict-free halves

typedef __attribute__((ext_vector_type(2))) float v2f;
typedef __attribute__((ext_vector_type(8))) float v8f;

// ---------------- helpers ----------------
__device__ __forceinline__ float block_reduce(float v, float* smem) {
  int lane = threadIdx.x & 31;
  int wid  = threadIdx.x >> 5;
  int nw   = (blockDim.x + 31) >> 5;
#pragma unroll
  for (int o = 16; o > 0; o >>= 1) v += __shfl_down(v, o, 32);
  if (lane == 0) smem[wid] = v;
  __syncthreads();
  float s = (threadIdx.x < nw) ? smem[threadIdx.x] : 0.f;
  if (wid == 0) {
#pragma unroll
    for (int o = 16; o > 0; o >>= 1) s += __shfl_down(s, o, 32);
    if (lane == 0) smem[0] = s;
  }
  __syncthreads();
  float r = smem[0];
  __syncthreads();
  return r;
}

__device__ __forceinline__ float silu_f(float v) {
  return v / (1.f + __expf(-v));
}
__device__ __forceinline__ float softplus_f(float v) {
  return (v > 20.f) ? v : log1pf(__expf(v));
}

// ---------------- kernel 1: input LayerNorm over C ----------------
__global__ void k_ln_in(const float* __restrict__ x, const float* __restrict__ w,
                        const float* __restrict__ b, float* __restrict__ hx) {
  __shared__ float smem[16];
  int bl = blockIdx.x;
  int c  = threadIdx.x;                  // blockDim = 192
  float v  = x[(size_t)bl * CC + c];
  float mu = block_reduce(v, smem) * (1.f / CC);
  float dvv = v - mu;
  float var = block_reduce(dvv * dvv, smem) * (1.f / CC);
  hx[(size_t)bl * CC + c] = dvv * rsqrtf(var + EPSL) * w[c] + b[c];
}

// =====================================================================
// Tiled WMMA f32 GEMM:  D[M,N] = A[M,K] @ Wt[N,K]^T (+resid)
// Block = 4 waves; block tile 128M x 64N; weight K-chunks staged in LDS.
// Per wave: two 16-row M tiles x four 16-col N tiles = 8 accumulators.
// Fragment layouts per CDNA5 ISA 7.12.2 (V_WMMA_F32_16X16X4_F32):
//   A 16x4: lane<16 -> (m=lane, K=k,k+1) ; lane>=16 -> (m=lane-16, K=k+2,k+3)
//   B 4x16: lane<16 -> (n=lane, K=k,k+1) ; lane>=16 -> (n=lane-16, K=k+2,k+3)
//   C/D:    vgpr v, lane t -> (m = v + (t>>4)*8, n = t&15)
// Requires: M % 128 == 0, N % 64 == 0, K % 192 == 0 (K=192 or 384 here).
// =====================================================================
__global__ __launch_bounds__(128) void k_gemm_tiled(
    const float* __restrict__ A, const float* __restrict__ Wt,
    const float* __restrict__ resid, float* __restrict__ D,
    int M, int N, int K) {
  __shared__ float lds[64 * LP];        // 49,664 B
  int tid  = threadIdx.x;
  int lane = tid & 31;
  int wave = tid >> 5;
  int nlo  = lane & 15;
  int half = lane >> 4;
  int n0 = blockIdx.x * 64;
  int m0 = blockIdx.y * 128 + wave * 16;
  int m1 = m0 + 64;

  v8f acc0[4], acc1[4];
#pragma unroll
  for (int t = 0; t < 4; ++t)
#pragma unroll
    for (int v = 0; v < 8; ++v) { acc0[t][v] = 0.f; acc1[t][v] = 0.f; }

  const float* arow0 = A + (size_t)(m0 + nlo) * K;
  const float* arow1 = A + (size_t)(m1 + nlo) * K;

  for (int kc0 = 0; kc0 < K; kc0 += KC) {
    __syncthreads();
    // cooperative stage of Wt[n0..n0+63, kc0..kc0+191] into LDS (v2f chunks)
    for (int idx = tid; idx < 64 * (KC / 2); idx += 128) {
      int row = idx / (KC / 2);
      int c2  = (idx - row * (KC / 2)) * 2;
      *(v2f*)&lds[row * LP + c2] =
          *(const v2f*)&Wt[(size_t)(n0 + row) * K + kc0 + c2];
    }
    __syncthreads();
#pragma unroll 4
    for (int k = 0; k < KC; k += 4) {
      int kb = k + half * 2;
      v2f af0 = *(const v2f*)(arow0 + kc0 + kb);
      v2f af1 = *(const v2f*)(arow1 + kc0 + kb);
#pragma unroll
      for (int t = 0; t < 4; ++t) {
        v2f bf = *(const v2f*)&lds[(t * 16 + nlo) * LP + kb];
        acc0[t] = __builtin_amdgcn_wmma_f32_16x16x4_f32(
            false, af0, false, bf, (short)0, acc0[t], false, false);
        acc1[t] = __builtin_amdgcn_wmma_f32_16x16x4_f32(
            false, af1, false, bf, (short)0, acc1[t], false, false);
      }
    }
  }

#pragma unroll
  for (int t = 0; t < 4; ++t) {
#pragma unroll
    for (int v = 0; v < 8; ++v) {
      int n  = n0 + t * 16 + nlo;
      int ma = m0 + v + half * 8;
      int mb = m1 + v + half * 8;
      float va = acc0[t][v];
      float vb = acc1[t][v];
      if (resid) {
        va += resid[(size_t)ma * N + n];
        vb += resid[(size_t)mb * N + n];
      }
      D[(size_t)ma * N + n] = va;
      D[(size_t)mb * N + n] = vb;
    }
  }
}

// ---------------- small WMMA GEMM (one wave, for x_proj: N=48) -------------
template<int NT>
__global__ void k_gemm_wmma(const float* __restrict__ A, const float* __restrict__ Wt,
                            float* __restrict__ D, int M, int N, int K) {
  int lane = threadIdx.x;                // 32 threads
  int nlo  = lane & 15;
  int half = lane >> 4;
  int m0 = blockIdx.y * 16;
  int n0 = blockIdx.x * (16 * NT);

  v8f acc[NT];
#pragma unroll
  for (int t = 0; t < NT; ++t)
#pragma unroll
    for (int v = 0; v < 8; ++v) acc[t][v] = 0.f;

  const float* arow = A + (size_t)(m0 + nlo) * K;
  for (int k = 0; k < K; k += 4) {
    int kb = k + half * 2;
    v2f af = *(const v2f*)(arow + kb);
#pragma unroll
    for (int t = 0; t < NT; ++t) {
      v2f bf = *(const v2f*)(Wt + (size_t)(n0 + t * 16 + nlo) * K + kb);
      acc[t] = __builtin_amdgcn_wmma_f32_16x16x4_f32(
          false, af, false, bf, (short)0, acc[t], false, false);
    }
  }
#pragma unroll
  for (int t = 0; t < NT; ++t)
#pragma unroll
    for (int v = 0; v < 8; ++v)
      D[(size_t)(m0 + v + half * 8) * N + n0 + t * 16 + nlo] = acc[t][v];
}

// ---------------- pad x_proj_w (44x384) -> (48x384) with zeros -------------
__global__ void k_pad_w(const float* __restrict__ src, float* __restrict__ dst) {
  int i = blockIdx.x * 256 + threadIdx.x;
  if (i < NPAD * DI) {
    int row = i / DI;
    dst[i] = (row < DR + 2 * NS) ? src[i] : 0.f;
  }
}

// ---------------- depthwise 3x3x3 conv + bias + SiLU -----------------------
__global__ void k_conv(const float* __restrict__ xz, const float* __restrict__ cw,
                       const float* __restrict__ cb, float* __restrict__ u) {
  int ch = threadIdx.x;                       // 384
  int w  = blockIdx.x & (WW - 1);
  int h  = blockIdx.x >> 5;                   // grid.x = HH*WW = 1024
  int dd = blockIdx.y;                        // 8
  int b  = blockIdx.z;                        // 4
  float acc = cb[ch];
  const float* wt = cw + ch * 27;
#pragma unroll
  for (int i = 0; i < 3; ++i) {
    int d2 = dd + i - 1; if (d2 < 0 || d2 >= DD) continue;
#pragma unroll
    for (int j = 0; j < 3; ++j) {
      int h2 = h + j - 1; if (h2 < 0 || h2 >= HH) continue;
#pragma unroll
      for (int k = 0; k < 3; ++k) {
        int w2 = w + k - 1; if (w2 < 0 || w2 >= WW) continue;
        size_t l2 = (size_t)(d2 * HH + h2) * WW + w2;
        acc += xz[((size_t)b * LL + l2) * (2 * DI) + ch] * wt[(i * 3 + j) * 3 + k];
      }
    }
  }
  size_t l = (size_t)(dd * HH + h) * WW + w;
  u[((size_t)b * LL + l) * DI + ch] = silu_f(acc);
}

// ---------------- delta = softplus(dts @ dt_w^T + dt_b) --------------------
__global__ void k_delta(const float* __restrict__ xdbl, const float* __restrict__ dtw,
                        const float* __restrict__ dtb, float* __restrict__ delta) {
  int d  = threadIdx.x;                      // 384
  int bl = blockIdx.x;
  const float* r = xdbl + (size_t)bl * NPAD; // cols 0..11 are dts
  float s = dtb[d];
#pragma unroll
  for (int j = 0; j < DR; ++j) s += r[j] * dtw[d * DR + j];
  delta[(size_t)bl * DI + d] = softplus_f(s);
}

// ---------------- scan phase 1: per-chunk decay product + local state ------
__global__ void k_scan1(const float* __restrict__ u, const float* __restrict__ delta,
                        const float* __restrict__ xdbl, const float* __restrict__ Alogs,
                        float* __restrict__ Pg, float* __restrict__ Qg) {
  int d = threadIdx.x;                       // 384
  int c = blockIdx.x;                        // chunk
  int b = blockIdx.y;
  float a[NS], P[NS], Q[NS];
#pragma unroll
  for (int n = 0; n < NS; ++n) {
    a[n] = -__expf(Alogs[d * NS + n]);
    P[n] = 1.f;  Q[n] = 0.f;
  }
  int l0 = c * CH;
  for (int l = l0; l < l0 + CH; ++l) {
    size_t row = (size_t)b * LL + l;
    float dl = delta[row * DI + d];
    float ul = u[row * DI + d];
    float xv = dl * ul;
    const float* xd = xdbl + row * NPAD;
#pragma unroll
    for (int n = 0; n < NS; ++n) {
      float dA = __expf(dl * a[n]);
      Q[n] = dA * Q[n] + xv * xd[DR + n];
      P[n] *= dA;
    }
  }
  size_t base = (((size_t)b * NCH + c) * NS) * DI + d;
#pragma unroll
  for (int n = 0; n < NS; ++n) {
    Pg[base + (size_t)n * DI] = P[n];
    Qg[base + (size_t)n * DI] = Q[n];
  }
}

// ---------------- scan phase 2: sequential carry across chunks -------------
__global__ void k_scan_fix(const float* __restrict__ Pg, const float* __restrict__ Qg,
                           float* __restrict__ Hinit) {
  int d = threadIdx.x;                       // 384
  int b = blockIdx.x;
  float h[NS];
#pragma unroll
  for (int n = 0; n < NS; ++n) h[n] = 0.f;
  for (int c = 0; c < NCH; ++c) {
    size_t base = (((size_t)b * NCH + c) * NS) * DI + d;
#pragma unroll
    for (int n = 0; n < NS; ++n) {
      Hinit[base + (size_t)n * DI] = h[n];
      h[n] = Pg[base + (size_t)n * DI] * h[n] + Qg[base + (size_t)n * DI];
    }
  }
}

// ---------------- scan phase 3: re-scan chunk with true init, emit y -------
__global__ void k_scan2(float* __restrict__ u, const float* __restrict__ delta,
                        const float* __restrict__ xdbl, const float* __restrict__ Alogs,
                        const float* __restrict__ Hinit, const float* __restrict__ Ds) {
  int d = threadIdx.x;
  int c = blockIdx.x;
  int b = blockIdx.y;
  float a[NS], h[NS];
  size_t base = (((size_t)b * NCH + c) * NS) * DI + d;
#pragma unroll
  for (int n = 0; n < NS; ++n) {
    a[n] = -__expf(Alogs[d * NS + n]);
    h[n] = Hinit[base + (size_t)n * DI];
  }
  float dsk = Ds[d];
  int l0 = c * CH;
  for (int l = l0; l < l0 + CH; ++l) {
    size_t row = (size_t)b * LL + l;
    float dl = delta[row * DI + d];
    float ul = u[row * DI + d];
    float xv = dl * ul;
    const float* xd = xdbl + row * NPAD;
    float y = 0.f;
#pragma unroll
    for (int n = 0; n < NS; ++n) {
      float dA = __expf(dl * a[n]);
      h[n] = dA * h[n] + xv * xd[DR + n];
      y += h[n] * xd[DR + NS + n];
    }
    u[row * DI + d] = y + dsk * ul;          // in-place: u no longer needed
  }
}

// ---------------- output LayerNorm over DI, gate with silu(z) --------------
__global__ void k_ln_gate(const float* __restrict__ y, const float* __restrict__ xz,
                          const float* __restrict__ ow, const float* __restrict__ ob,
                          float* __restrict__ g) {
  __shared__ float smem[16];
  int bl = blockIdx.x;
  int d  = threadIdx.x;                      // 384
  float v  = y[(size_t)bl * DI + d];
  float mu = block_reduce(v, smem) * (1.f / DI);
  float dv = v - mu;
  float var = block_reduce(dv * dv, smem) * (1.f / DI);
  float ln = dv * rsqrtf(var + EPSL) * ow[d] + ob[d];
  float z  = xz[(size_t)bl * (2 * DI) + DI + d];
  g[(size_t)bl * DI + d] = ln * silu_f(z);
}

// ---------------- launch ----------------
extern "C" void kernel_launch(void* const* d_in, const int* in_sizes, int n_in,
                              void* d_out, int out_size, void* d_ws, size_t ws_size,
                              hipStream_t stream) {
  (void)in_sizes; (void)n_in; (void)out_size; (void)ws_size;
  const float* x     = (const float*)d_in[0];
  const float* ln1w  = (const float*)d_in[1];
  const float* ln1b  = (const float*)d_in[2];
  const float* Win   = (const float*)d_in[3];   // (768,192)
  const float* convw = (const float*)d_in[4];   // (384,1,3,3,3)
  const float* convb = (const float*)d_in[5];
  const float* xprjw = (const float*)d_in[6];   // (44,384)
  const float* dtw   = (const float*)d_in[7];   // (384,12)
  const float* dtb   = (const float*)d_in[8];
  const float* Alogs = (const float*)d_in[9];   // (384,16)
  const float* Dsk   = (const float*)d_in[10];
  const float* onw   = (const float*)d_in[11];
  const float* onb   = (const float*)d_in[12];
  const float* Wout  = (const float*)d_in[13];  // (192,384)
  float* out = (float*)d_out;

  float* ws = (float*)d_ws;
  size_t oHX = 0;                              // (BL,192)
  size_t oXZ = oHX + (size_t)BL * CC;          // (BL,768)
  size_t oU  = oXZ + (size_t)BL * 2 * DI;      // (BL,384) u, then y in place
  size_t oXD = oU  + (size_t)BL * DI;          // (BL,48)
  size_t oWP = oXD + (size_t)BL * NPAD;        // (48,384)
  size_t oDL = oWP + (size_t)NPAD * DI;        // (BL,384) delta, then g
  size_t oP  = oDL + (size_t)BL * DI;
  size_t oQ  = oP  + (size_t)BB * NCH * NS * DI;
  size_t oHI = oQ  + (size_t)BB * NCH * NS * DI;

  float* hx   = ws + oHX;
  float* xz   = ws + oXZ;
  float* u    = ws + oU;
  float* xdbl = ws + oXD;
  float* wpad = ws + oWP;
  float* dl   = ws + oDL;   // delta / gate buffer
  float* Pg   = ws + oP;
  float* Qg   = ws + oQ;
  float* Hi   = ws + oHI;

  // 1. LN over C
  k_ln_in<<<BL, CC, 0, stream>>>(x, ln1w, ln1b, hx);
  // 2. xz = hx @ W_in^T : M=32768, N=768, K=192 (LDS-tiled WMMA)
  k_gemm_tiled<<<dim3((2 * DI) / 64, BL / 128), 128, 0, stream>>>(
      hx, Win, nullptr, xz, BL, 2 * DI, CC);
  // 3. pad x_proj weights
  k_pad_w<<<(NPAD * DI + 255) / 256, 256, 0, stream>>>(xprjw, wpad);
  // 4. depthwise conv3d + SiLU -> u
  k_conv<<<dim3(HH * WW, DD, BB), DI, 0, stream>>>(xz, convw, convb, u);
  // 5. x_dbl = u @ x_proj^T : M=32768, N=48, K=384 (small WMMA)
  k_gemm_wmma<3><<<dim3(1, BL / 16), 32, 0, stream>>>(
      u, wpad, xdbl, BL, NPAD, DI);
  // 6. delta
  k_delta<<<BL, DI, 0, stream>>>(xdbl, dtw, dtb, dl);
  // 7-9. chunked selective scan
  k_scan1<<<dim3(NCH, BB), DI, 0, stream>>>(u, dl, xdbl, Alogs, Pg, Qg);
  k_scan_fix<<<BB, DI, 0, stream>>>(Pg, Qg, Hi);
  k_scan2<<<dim3(NCH, BB), DI, 0, stream>>>(u, dl, xdbl, Alogs, Hi, Dsk);
  // 10. LN over DI + silu(z) gate -> g (reuse delta buffer)
  k_ln_gate<<<BL, DI, 0, stream>>>(u, xz, onw, onb, dl);
  // 11. out = x + g @ W_out^T : M=32768, N=192, K=384 (LDS-tiled WMMA)
  k_gemm_tiled<<<dim3(CC / 64, BL / 128), 128, 0, stream>>>(
      dl, Wout, x, out, BL, CC, DI);
}